// RelativePositionAttention_22565758173249
// MI455X (gfx1250) — compile-verified
//
#include <hip/hip_runtime.h>

// CDNA5 / gfx1250, wave32. Reference collapses algebraically:
//   softmax over w of (Q[t] - K[w]) drops Q (constant shift over w), and
//   sum_w softmax(...) == 1, so out == Vh and the whole net is
//       output = values @ (Wv^T permuted) @ Wp^T == values @ M,  M: 256x256.
// Pipeline: Bp = perm(Wp^T)  ->  M = Wv^T x Bp  ->  out = values x M.
// Full fp32 via V_WMMA_F32_16X16X4_F32; one wave per 16x16 C tile.
// All dims are compile-time constants so loads use immediate offsets.

typedef __attribute__((ext_vector_type(2))) float v2f;
typedef __attribute__((ext_vector_type(8))) float v8f;

static __device__ __forceinline__ v8f wmma4(v2f a, v2f b, v8f c) {
    return __builtin_amdgcn_wmma_f32_16x16x4_f32(false, a, false, b,
                                                 (short)0, c, false, false);
}

// Bp[j][e] = Wp[e][(j%256)*4 + j/256]   (Wp is [256][1024] row-major)
__global__ void wp_permute_kernel(const float* __restrict__ Wp,
                                  float* __restrict__ Bp) {
    int idx = blockIdx.x * 256 + threadIdx.x;   // 0 .. 1024*256-1
    int e = idx & 255;
    int j = idx >> 8;
    Bp[idx] = Wp[e * 1024 + ((j & 255) << 2) + (j >> 8)];
}

// C[M][256] = A x B. B layout [K][256]. A layout: AT ? [K][256] (A^T stored)
//                                                    : [M][K] row-major.
// One wave per 16x16 C tile; N is always 256 -> 16 tile columns.
template <int K, bool AT>
__global__ void gemm_f32_wmma(const float* __restrict__ A,
                              const float* __restrict__ B,
                              float* __restrict__ C) {
    constexpr int LD = 256;
    const int lane = threadIdx.x & 31;
    const int wave = (blockIdx.x * blockDim.x + threadIdx.x) >> 5;
    const int tm = wave >> 4;        // tile row
    const int tn = wave & 15;        // tile col
    const int half = lane >> 4;      // lanes 16-31 hold K = k0+2, k0+3
    const int l16  = lane & 15;

    const float* Ap = AT ? (A + (2 * half) * LD + tm * 16 + l16)
                         : (A + (tm * 16 + l16) * (AT ? LD : K) + 2 * half);
    const float* Bq = B + (2 * half) * LD + tn * 16 + l16;

    v8f c0 = {}, c1 = {};
#pragma unroll 4
    for (int kb = 0; kb < K / 4; kb += 2) {     // uniform trip count
        v2f a0, a1, b0, b1;
        if (AT) {
            a0.x = Ap[0];          a0.y = Ap[LD];
            a1.x = Ap[4 * LD];     a1.y = Ap[5 * LD];
        } else {
            a0 = *(const v2f*)(Ap);             // k0,k0+1 contiguous: b64 load
            a1 = *(const v2f*)(Ap + 4);
        }
        b0.x = Bq[0];          b0.y = Bq[LD];
        b1.x = Bq[4 * LD];     b1.y = Bq[5 * LD];
        c0 = wmma4(a0, b0, c0);
        c1 = wmma4(a1, b1, c1);
        Ap += AT ? 8 * LD : 8;
        Bq += 8 * LD;
    }
    const v8f c = c0 + c1;
    const int row = tm * 16 + 8 * half;   // C layout: VGPR r -> M = r + 8*half
    const int col = tn * 16 + l16;
#pragma unroll
    for (int r = 0; r < 8; ++r)
        C[(row + r) * LD + col] = c[r];
}

extern "C" void kernel_launch(void* const* d_in, const int* in_sizes, int n_in,
                              void* d_out, int out_size, void* d_ws, size_t ws_size,
                              hipStream_t stream) {
    // setup_inputs order:
    // 0: position_embeddings (dead), 1: values [2,256,256],
    // 2: ln_w (dead), 3: ln_b (dead), 4: Wq (dead), 5: Wk (dead),
    // 6: Wv [1024,256], 7: Wp [256,1024]
    const float* values = (const float*)d_in[1];
    const float* Wv     = (const float*)d_in[6];
    const float* Wp     = (const float*)d_in[7];
    float* out = (float*)d_out;                 // [512, 256]

    float* Bp = (float*)d_ws;                   // [1024][256]  (1 MB)
    float* Mm = Bp + 1024 * 256;                // [256][256]   (256 KB)

    // 1) Bp = permuted Wp^T  (coalesced writes)
    wp_permute_kernel<<<1024, 256, 0, stream>>>(Wp, Bp);

    // 2) Mm[i][e] = sum_j Wv[j][i] * Bp[j][e]   (M=256, N=256, K=1024)
    //    256 tiles -> 256 waves -> 32 blocks of 8 waves
    gemm_f32_wmma<1024, true><<<32, 256, 0, stream>>>(Wv, Bp, Mm);

    // 3) out[r][e] = sum_i values[r][i] * Mm[i][e]  (M=512, N=256, K=256)
    //    512 tiles -> 512 waves -> 64 blocks of 8 waves
    gemm_f32_wmma<256, false><<<64, 256, 0, stream>>>(values, Mm, out);
}